// histogram_binning_posterior_estimator_79431125172851
// MI455X (gfx1250) — compile-verified
//
#include <hip/hip_runtime.h>
#include <stdint.h>

// Histogram-binning posterior estimator, MI455X (gfx1250), wave32.
//
// Roofline: 262 MB in + 262 MB out @ 23.3 TB/s ~= 22.5 us floor; compute is
// ~131M v_exp_f32 -> fully memory bound. Strategy: one wave per row, stage the
// row into LDS with the CDNA5 async-DMA path (global_load_async_to_lds_b128,
// ASYNCcnt), read it into VGPRs once, do max/argmax + sum-exp + rescale from
// registers, and stream the result out with non-temporal 16B stores so the
// output doesn't evict the input stream from L2.

typedef float v4f __attribute__((ext_vector_type(4)));

#define CLS     1000            // classes per row
#define CHUNKS  250             // float4 chunks per row (1000/4)
#define WPB     8               // waves per block (256 threads, 8 rows/block)
#define LDS_V4  256             // float4 slots per wave region (4 KB, padded)

__global__ __launch_bounds__(256) void hbpe_kernel(
    const float* __restrict__ logits,
    const float* __restrict__ hist,
    float* __restrict__ out,
    int nRows, int nBins)
{
    __shared__ v4f smem[WPB * LDS_V4];          // 32 KB / block

    const int lane = threadIdx.x & 31;
    const int wave = threadIdx.x >> 5;
    const int row  = blockIdx.x * WPB + wave;
    if (row >= nRows) return;                   // wave-uniform exit

    const char* rowBase = (const char*)(logits + (size_t)row * CLS);
    v4f* myLds = smem + wave * LDS_V4;
    // generic->LDS byte offset: low 32 bits of a generic LDS address are the
    // LDS offset.
    unsigned ldsBase = (unsigned)(uintptr_t)myLds;

    // ---- async DMA: global -> LDS, tracked by ASYNCcnt --------------------
    #pragma unroll
    for (int k = 0; k < 8; ++k) {
        int chunk = k * 32 + lane;
        if (chunk < CHUNKS) {
            unsigned lds_addr = ldsBase + (unsigned)(chunk * 16);
            unsigned long long gaddr =
                (unsigned long long)(uintptr_t)rowBase +
                (unsigned long long)(chunk * 16);
            asm volatile("global_load_async_to_lds_b128 %0, %1, off"
                         :: "v"(lds_addr), "v"(gaddr)
                         : "memory");
        }
    }
    asm volatile("s_wait_asynccnt 0" ::: "memory");

    // ---- pull row into registers once (8 x float4 per lane) ---------------
    v4f vals[8];
    #pragma unroll
    for (int k = 0; k < 8; ++k) {
        int chunk = k * 32 + lane;
        v4f ninf = { -__builtin_inff(), -__builtin_inff(),
                     -__builtin_inff(), -__builtin_inff() };
        vals[k] = (chunk < CHUNKS) ? myLds[chunk] : ninf;
    }

    // ---- pass 1: row max + first-occurrence argmax ------------------------
    float m = -__builtin_inff();
    int   midx = 0x7fffffff;
    #pragma unroll
    for (int k = 0; k < 8; ++k) {
        int base = (k * 32 + lane) * 4;
        #pragma unroll
        for (int j = 0; j < 4; ++j) {
            float v = vals[k][j];
            if (v > m) { m = v; midx = base + j; }  // strict > keeps first idx
        }
    }
    #pragma unroll
    for (int off = 16; off >= 1; off >>= 1) {
        float om = __shfl_xor(m, off, 32);
        int   oi = __shfl_xor(midx, off, 32);
        if (om > m || (om == m && oi < midx)) { m = om; midx = oi; }
    }

    // ---- pass 2: S = sum exp(l - m)  (conf = 1/S) -------------------------
    float s = 0.0f;
    #pragma unroll
    for (int k = 0; k < 8; ++k) {
        #pragma unroll
        for (int j = 0; j < 4; ++j)
            s += __expf(vals[k][j] - m);            // exp(-inf)=0 for tail
    }
    #pragma unroll
    for (int off = 16; off >= 1; off >>= 1)
        s += __shfl_xor(s, off, 32);

    float conf = 1.0f / s;
    int bin = (int)(conf * (float)nBins);           // trunc, matches int32 cast
    bin = bin < 0 ? 0 : (bin > nBins - 1 ? nBins - 1 : bin);
    float post = hist[bin];
    if (post == -1.0f) post = conf;                 // empty-bin fallback
    // rem_sum = (S-1)/S; rescale factor for non-pred classes:
    // out_i = exp(l_i - m) * (1-post)/(S-1); out_pred = post; sum == 1.
    float scale = (1.0f - post) / (s - 1.0f);

    // ---- pass 3: write calibrated distribution (non-temporal) -------------
    float* outRow = out + (size_t)row * CLS;
    #pragma unroll
    for (int k = 0; k < 8; ++k) {
        int chunk = k * 32 + lane;
        if (chunk < CHUNKS) {
            v4f r;
            #pragma unroll
            for (int j = 0; j < 4; ++j) {
                int idx = chunk * 4 + j;
                float e = __expf(vals[k][j] - m) * scale;
                r[j] = (idx == midx) ? post : e;
            }
            __builtin_nontemporal_store(r, (v4f*)(outRow + chunk * 4));
        }
    }
}

extern "C" void kernel_launch(void* const* d_in, const int* in_sizes, int n_in,
                              void* d_out, int out_size, void* d_ws, size_t ws_size,
                              hipStream_t stream) {
    const float* logits = (const float*)d_in[0];   // (N, 1000) f32
    const float* hist   = (const float*)d_in[1];   // (15,) f32
    float* out = (float*)d_out;                    // (N, 1000) f32

    int nRows = in_sizes[0] / CLS;
    int nBins = in_sizes[1];
    int blocks = (nRows + WPB - 1) / WPB;

    hipLaunchKernelGGL(hbpe_kernel, dim3(blocks), dim3(WPB * 32), 0, stream,
                       logits, hist, out, nRows, nBins);
}